// Model_39676907882930
// MI455X (gfx1250) — compile-verified
//
#include <hip/hip_runtime.h>
#include <hip/hip_bf16.h>

typedef __attribute__((ext_vector_type(2))) float v2f;
typedef __attribute__((ext_vector_type(8))) float v8f;

// One wave32, one workgroup. Four V_WMMA_F32_16X16X4_F32 ops compute every
// matmul in the reference exactly (f32 in, f32 accumulate, RNE), with the
// 4x2 / 2x4 operands zero-padded into the 16x4 A / 4x16 B tiles per the
// CDNA5 VGPR layouts. Outputs are scattered straight from the D registers.
__global__ __launch_bounds__(32) void model_wmma_kernel(
    const float* __restrict__ x1, const float* __restrict__ x2,
    float* __restrict__ out) {
  const int lane = threadIdx.x;   // 0..31, one wave
  const int mlo  = lane & 15;     // row/col index within 16-lane half
  const int khi  = lane >> 4;     // 0 -> K={0,1}, 1 -> K={2,3}

  // Tiny inputs: every lane holds both matrices in registers.
  float a[8], b[8];
#pragma unroll
  for (int i = 0; i < 8; ++i) { a[i] = x1[i]; b[i] = x2[i]; }
  // a(i,j) = a[i*2+j]  (4x2 "a" = x1.reshape(4,2)); likewise b.

  // ---- A-operands (16x4 tiles, zero padded) -------------------------------
  // A = a (4x2): A[m][k], m<4, k<2  ->  lane m (khi==0), vgpr j holds k=j.
  v2f Aa = {0.f, 0.f};
  if (khi == 0 && mlo < 4) { Aa.x = a[mlo * 2 + 0]; Aa.y = a[mlo * 2 + 1]; }
  // A = a^T (2x4): A[m][k] = a[k][m], m<2, k<4 -> vgpr j holds k=2*khi+j.
  v2f AaT = {0.f, 0.f};
  if (mlo < 2) { AaT.x = a[(2 * khi + 0) * 2 + mlo]; AaT.y = a[(2 * khi + 1) * 2 + mlo]; }
  // A = b^T (2x4)
  v2f AbT = {0.f, 0.f};
  if (mlo < 2) { AbT.x = b[(2 * khi + 0) * 2 + mlo]; AbT.y = b[(2 * khi + 1) * 2 + mlo]; }

  // ---- B-operands (4x16 tiles, zero padded) -------------------------------
  // B = b^T (2x4): B[k][n] = b[n][k], k<2, n<4 -> lane n (khi==0), vgpr j = k.
  v2f BbT = {0.f, 0.f};
  if (khi == 0 && mlo < 4) { BbT.x = b[mlo * 2 + 0]; BbT.y = b[mlo * 2 + 1]; }
  // B = a^T (2x4): B[k][n] = a[n][k]
  v2f BaT = {0.f, 0.f};
  if (khi == 0 && mlo < 4) { BaT.x = a[mlo * 2 + 0]; BaT.y = a[mlo * 2 + 1]; }
  // B = a (4x2): B[k][n] = a[k][n], k<4, n<2 -> vgpr j holds k=2*khi+j.
  // Identical element mapping to AaT (same lane/vgpr indexing).
  v2f Ba = AaT;

  // ---- Matrix ops: D = A x B + 0 on the CDNA5 WMMA pipe -------------------
  v8f zero = {};
  // (neg_a, A, neg_b, B, c_mod, C, reuse_a, reuse_b)
  v8f d1 = __builtin_amdgcn_wmma_f32_16x16x4_f32(false, Aa,  false, BbT, (short)0, zero, false, false); // a @ b^T  -> x3, x4, v6
  v8f d2 = __builtin_amdgcn_wmma_f32_16x16x4_f32(false, Aa,  false, BaT, (short)0, zero, false, false); // a @ a^T  -> v4
  v8f d3 = __builtin_amdgcn_wmma_f32_16x16x4_f32(false, AaT, false, Ba,  (short)0, zero, false, false); // a^T @ a  -> v3
  v8f d4 = __builtin_amdgcn_wmma_f32_16x16x4_f32(false, AbT, false, Ba,  (short)0, zero, false, false); // b^T @ a  -> v5

  // ---- Scatter results. D layout: D[m][n] -> lane n (m<8), vgpr m. --------
  // Output concatenation (f32):
  //   x3[16]@0, x4[16]@16, v1[8]@32, v2[8]@40, v3[4]@48,
  //   v4[16]@52, v5[4]@68, v6[16]@72, v7[8]@88   (total 96)
  if (khi == 0) {
    const int n = mlo;
    if (n < 4) {
#pragma unroll
      for (int m = 0; m < 4; ++m) {
        const float v = d1[m];
        out[ 0 + m * 4 + n] = v;      // x3 = a @ b^T
        out[16 + n * 4 + m] = v;      // x4 = x3^T
        out[72 + m * 4 + n] = v;      // v6 = x3
        out[52 + m * 4 + n] = d2[m];  // v4 = a @ a^T
      }
    }
    if (n < 2) {
#pragma unroll
      for (int m = 0; m < 2; ++m) {
        out[48 + m * 2 + n] = d3[m];  // v3 = a^T @ a (2x2)
        out[68 + m * 2 + n] = d4[m];  // v5 = b^T @ a (2x2)
      }
    }
  }
  // Pure transposes / copies of the input, written from registers.
  if (lane < 8) {
    out[32 + lane] = a[lane];                          // v1 = a (flat == x1 flat)
    const float t = a[(lane & 3) * 2 + (lane >> 2)];   // a^T flat: [j*4+i] = a[i][j]
    out[40 + lane] = t;                                // v2 = a^T
    out[88 + lane] = t;                                // v7 = a^T
  }
}

extern "C" void kernel_launch(void* const* d_in, const int* in_sizes, int n_in,
                              void* d_out, int out_size, void* d_ws, size_t ws_size,
                              hipStream_t stream) {
  (void)in_sizes; (void)n_in; (void)out_size; (void)d_ws; (void)ws_size;
  const float* x1 = (const float*)d_in[0];
  const float* x2 = (const float*)d_in[1];
  float* out = (float*)d_out;
  model_wmma_kernel<<<dim3(1), dim3(32), 0, stream>>>(x1, x2, out);
}